// LearnedIGCN_67095979098484
// MI455X (gfx1250) — compile-verified
//
#include <hip/hip_runtime.h>
#include <cmath>

typedef __attribute__((ext_vector_type(2))) float v2f;
typedef __attribute__((ext_vector_type(8))) float v8f;

#define N_NODES 100000
#define DIM     256
#define CLS     40
#define NE      3200000
#define N_IDS   1000
#define NC      (N_NODES * CLS)      // 4,000,000
#define NC4     (NC / 4)             // 1,000,000
#define MAXIT   16
#define TOL2    1e-4f

// ---- workspace layout (float offsets) ----
// scal[0..16]  = rr[t]   (rr[0] = <b,b>)
// scal[32+t]   = pq[t]
#define SCAL_OFF 0
#define WPAD_OFF 64                  // 256*48 padded W
#define X_OFF    16384
#define R_OFF    (X_OFF + NC)
#define P_OFF    (R_OFF + NC)
#define Y_OFF    (P_OFF + NC)

__device__ __forceinline__ float one_minus_eps(const float* e0) {
  return 1.0f - 1.0f / (1.0f + __expf(-e0[0]));
}

__device__ __forceinline__ bool cg_active(const float* s, int t) {
  // iteration t (1-based) runs while rr[t-1] > tol^2 * rr[0]
  return s[t - 1] > TOL2 * s[0];
}

__device__ __forceinline__ void block_reduce_atomic(float local, float* slot) {
  __shared__ float red[256];
  red[threadIdx.x] = local;
  __syncthreads();
  for (int off = 128; off > 0; off >>= 1) {
    if ((int)threadIdx.x < off) red[threadIdx.x] += red[threadIdx.x + off];
    __syncthreads();
  }
  if (threadIdx.x == 0) unsafeAtomicAdd(slot, red[0]);
}

// ---- init: zero solution vector and all scalar slots ----
__global__ void k_init(float* __restrict__ x, float* __restrict__ scal) {
  int i = blockIdx.x * blockDim.x + threadIdx.x;
  if (i < NC4) ((float4*)x)[i] = make_float4(0.f, 0.f, 0.f, 0.f);
  if (i < 64) scal[i] = 0.f;
}

// ---- pad W [256x40] -> [256x48] (zero-fill cols 40..47) ----
__global__ void k_wpad(const float* __restrict__ W, float* __restrict__ wpad) {
  int i = blockIdx.x * blockDim.x + threadIdx.x;
  if (i >= 256 * 48) return;
  int k = i / 48, c = i - k * 48;
  wpad[i] = (c < CLS) ? W[k * CLS + c] : 0.f;
}

// ---- GEMM: r = p = F @ W + b  via V_WMMA_F32_16X16X4_F32 ----
// one wave per 16x16 output tile; grid.y = column tile (3 tiles of 16, cols>=40 dropped)
__global__ void k_gemm(const float* __restrict__ F, const float* __restrict__ bias,
                       const float* __restrict__ wpad,
                       float* __restrict__ r, float* __restrict__ p) {
  __shared__ float lw[256 * 16];      // W slice: [k=0..255][n=0..15] for this col tile
  const int ct = blockIdx.y;
  for (int i = threadIdx.x; i < 256 * 16; i += blockDim.x) {
    int k = i >> 4, n = i & 15;
    lw[i] = wpad[k * 48 + ct * 16 + n];
  }
  __syncthreads();

  const int wave = threadIdx.x >> 5;
  const int lane = threadIdx.x & 31;
  const int rowTile = blockIdx.x * 8 + wave;
  if (rowTile >= N_NODES / 16) return;

  const int m  = lane & 15;           // A: row M per lane; B/D: col N per lane
  const int kh = lane >> 4;           // lane halves split K (ISA 7.12.2, 32-bit A 16x4)
  const float* arow = F + (size_t)(rowTile * 16 + m) * DIM + 2 * kh;

  v8f acc = {};
  for (int k0 = 0; k0 < DIM; k0 += 4) {
    v2f a, b;
    a.x = arow[k0];
    a.y = arow[k0 + 1];
    b.x = lw[(k0 + 2 * kh) * 16 + m];
    b.y = lw[(k0 + 2 * kh + 1) * 16 + m];
    acc = __builtin_amdgcn_wmma_f32_16x16x4_f32(false, a, false, b,
                                                (short)0, acc, false, false);
  }

  const int col = ct * 16 + m;
  if (col < CLS) {
    float bv = bias[col];
    #pragma unroll
    for (int i = 0; i < 8; ++i) {     // C/D: VGPR i -> M=i (lanes 0-15) / M=i+8 (16-31)
      int row = rowTile * 16 + ((lane < 16) ? i : (i + 8));
      float v = acc[i] + bv;
      r[row * CLS + col] = v;
      p[row * CLS + col] = v;
    }
  }
}

// ---- rr0 = <r, r> ----
__global__ void k_dot_rr0(const float* __restrict__ r, float* scal) {
  int i = blockIdx.x * blockDim.x + threadIdx.x;
  float local = 0.f;
  if (i < NC4) {
    float4 v = ((const float4*)r)[i];
    local = v.x * v.x + v.y * v.y + v.z * v.z + v.w * v.w;
  }
  block_reduce_atomic(local, &scal[0]);
}

__global__ void k_zero_y(float* __restrict__ y) {
  int i = blockIdx.x * blockDim.x + threadIdx.x;
  if (i < NC4) ((float4*)y)[i] = make_float4(0.f, 0.f, 0.f, 0.f);
}

// ---- SpMV scatter: y[row] += data * p[col], float4 chunk per thread ----
__global__ void k_spmv(const float* __restrict__ scal, int t,
                       const float* __restrict__ data, const int* __restrict__ row,
                       const int* __restrict__ col,
                       const float* __restrict__ p, float* __restrict__ y) {
  if (!cg_active(scal, t)) return;
  int tid = blockIdx.x * blockDim.x + threadIdx.x;   // NE*10 = 32M threads exactly
  if (tid >= NE * 10) return;
  int e = tid / 10;
  int q = tid - e * 10;
  int c0 = q * 4;
  float d = data[e];
  int cidx = col[e] * CLS + c0;
  int ridx = row[e] * CLS + c0;
  float4 v = *(const float4*)(p + cidx);
  unsafeAtomicAdd(y + ridx + 0, d * v.x);
  unsafeAtomicAdd(y + ridx + 1, d * v.y);
  unsafeAtomicAdd(y + ridx + 2, d * v.z);
  unsafeAtomicAdd(y + ridx + 3, d * v.w);
}

// ---- pq[t] = <p, q> with q = p - (1-eps)*y ----
__global__ void k_dot_pq(float* scal, int t, const float* __restrict__ e0,
                         const float* __restrict__ p, const float* __restrict__ y) {
  bool active = cg_active(scal, t);
  float local = 0.f;
  if (active) {
    float ome = one_minus_eps(e0);
    int i = blockIdx.x * blockDim.x + threadIdx.x;
    if (i < NC4) {
      float4 pv = ((const float4*)p)[i];
      float4 yv = ((const float4*)y)[i];
      float4 qv = make_float4(pv.x - ome * yv.x, pv.y - ome * yv.y,
                              pv.z - ome * yv.z, pv.w - ome * yv.w);
      local = pv.x * qv.x + pv.y * qv.y + pv.z * qv.z + pv.w * qv.w;
    }
  }
  if (active) block_reduce_atomic(local, &scal[32 + t]);
}

// ---- x += alpha p; r -= alpha q; rr[t] = <r,r> ----
__global__ void k_update(float* scal, int t, const float* __restrict__ e0,
                         const float* __restrict__ p, const float* __restrict__ y,
                         float* __restrict__ x, float* __restrict__ r) {
  bool active = cg_active(scal, t);
  float local = 0.f;
  if (active) {
    float ome = one_minus_eps(e0);
    float alpha = scal[t - 1] / scal[32 + t];
    int i = blockIdx.x * blockDim.x + threadIdx.x;
    if (i < NC4) {
      float4 pv = ((const float4*)p)[i];
      float4 yv = ((const float4*)y)[i];
      float4 xv = ((float4*)x)[i];
      float4 rv = ((float4*)r)[i];
      xv.x += alpha * pv.x; xv.y += alpha * pv.y;
      xv.z += alpha * pv.z; xv.w += alpha * pv.w;
      rv.x -= alpha * (pv.x - ome * yv.x);
      rv.y -= alpha * (pv.y - ome * yv.y);
      rv.z -= alpha * (pv.z - ome * yv.z);
      rv.w -= alpha * (pv.w - ome * yv.w);
      ((float4*)x)[i] = xv;
      ((float4*)r)[i] = rv;
      local = rv.x * rv.x + rv.y * rv.y + rv.z * rv.z + rv.w * rv.w;
    }
  }
  if (active) block_reduce_atomic(local, &scal[t]);
}

// ---- p = r + beta p ----
__global__ void k_pupdate(const float* __restrict__ scal, int t,
                          const float* __restrict__ r, float* __restrict__ p) {
  if (!cg_active(scal, t)) return;
  float beta = scal[t] / scal[t - 1];
  int i = blockIdx.x * blockDim.x + threadIdx.x;
  if (i < NC4) {
    float4 rv = ((const float4*)r)[i];
    float4 pv = ((float4*)p)[i];
    pv.x = rv.x + beta * pv.x; pv.y = rv.y + beta * pv.y;
    pv.z = rv.z + beta * pv.z; pv.w = rv.w + beta * pv.w;
    ((float4*)p)[i] = pv;
  }
}

// ---- out = x[ids] ----
__global__ void k_gather(const float* __restrict__ x, const int* __restrict__ ids,
                         float* __restrict__ out) {
  int tid = blockIdx.x * blockDim.x + threadIdx.x;
  if (tid >= N_IDS * CLS) return;
  int i = tid / CLS, c = tid - i * CLS;
  out[tid] = x[ids[i] * CLS + c];
}

extern "C" void kernel_launch(void* const* d_in, const int* in_sizes, int n_in,
                              void* d_out, int out_size, void* d_ws, size_t ws_size,
                              hipStream_t stream) {
  const float* F    = (const float*)d_in[0];
  const float* W    = (const float*)d_in[1];
  const float* bias = (const float*)d_in[2];
  const float* e0   = (const float*)d_in[3];
  const float* data = (const float*)d_in[4];
  const int*   row  = (const int*)d_in[5];
  const int*   col  = (const int*)d_in[6];
  const int*   ids  = (const int*)d_in[7];

  float* ws   = (float*)d_ws;
  float* scal = ws + SCAL_OFF;
  float* wpad = ws + WPAD_OFF;
  float* x    = ws + X_OFF;
  float* r    = ws + R_OFF;
  float* p    = ws + P_OFF;
  float* y    = ws + Y_OFF;

  const int VB = (NC4 + 255) / 256;        // 3907 blocks for 4M-float vector ops
  const int SB = (NE * 10) / 256;          // 125000 blocks, exact

  k_init<<<VB, 256, 0, stream>>>(x, scal);
  k_wpad<<<(256 * 48 + 255) / 256, 256, 0, stream>>>(W, wpad);
  k_gemm<<<dim3((N_NODES / 16 + 7) / 8, 3), 256, 0, stream>>>(F, bias, wpad, r, p);
  k_dot_rr0<<<VB, 256, 0, stream>>>(r, scal);

  for (int t = 1; t <= MAXIT; ++t) {
    k_zero_y<<<VB, 256, 0, stream>>>(y);
    k_spmv<<<SB, 256, 0, stream>>>(scal, t, data, row, col, p, y);
    k_dot_pq<<<VB, 256, 0, stream>>>(scal, t, e0, p, y);
    k_update<<<VB, 256, 0, stream>>>(scal, t, e0, p, y, x, r);
    k_pupdate<<<VB, 256, 0, stream>>>(scal, t, r, p);
  }

  k_gather<<<(N_IDS * CLS + 255) / 256, 256, 0, stream>>>(x, ids, (float*)d_out);
}